// EdgeConv_4561255268381
// MI455X (gfx1250) — compile-verified
//
#include <hip/hip_runtime.h>
#include <math.h>

typedef float v2f __attribute__((ext_vector_type(2)));
typedef float v8f __attribute__((ext_vector_type(8)));

#define B_ 4
#define N_ 512
#define C_ 64
#define O_ 64

// ---------------------------------------------------------------------------
// Kernel 1: per-batch GEMMs  a = x @ Wa, d = x @ Wb  via V_WMMA_F32_16X16X4_F32
//   writes u = a - d + bias  and  d  to workspace.
// One wave owns one 16x16 output tile and accumulates over K=64 in steps of 4.
// Fragment layouts per CDNA5 ISA 7.12.2:
//   A (16x4 f32):  lane(r = lane&15, h = lane>>4), vgpr j -> A[r][k0 + 2h + j]
//   B (4x16 f32):  lane(r,h), vgpr j              -> B[k0 + 2h + j][r]
//   C/D (16x16):   lane(r,h), vgpr i              -> D[i + 8h][r]
// ---------------------------------------------------------------------------
__global__ __launch_bounds__(128) void edgeconv_gemm(
    const float* __restrict__ x,     // (B,N,C)
    const float* __restrict__ W,     // (2C,O) row-major; Wa = rows [0,C), Wb = rows [C,2C)
    const float* __restrict__ bias,  // (O)
    float* __restrict__ u,           // ws: (B,N,O) = a - d + bias
    float* __restrict__ dws)         // ws: (B,N,O) = d
{
    const int wave = threadIdx.x >> 5;   // 0..3 -> N(=O) tile
    const int lane = threadIdx.x & 31;
    const int r    = lane & 15;
    const int h    = lane >> 4;
    const int b    = blockIdx.x >> 5;    // batch
    const int mt   = blockIdx.x & 31;    // M tile (rows of x)
    const int o    = wave * 16 + r;      // output channel for B/C/D columns
    const int rowBase = mt * 16;

    const float* xrow = x + (size_t)(b * N_ + rowBase + r) * C_;

    v8f accA = {};
    v8f accD = {};
    #pragma unroll
    for (int k0 = 0; k0 < C_; k0 += 4) {
        const int ka = k0 + 2 * h;
        v2f aF = *(const v2f*)(xrow + ka);           // A[r][ka], A[r][ka+1]
        v2f bA, bD;
        bA.x = W[(size_t)(ka)         * O_ + o];     // Wa[ka][o]
        bA.y = W[(size_t)(ka + 1)     * O_ + o];
        bD.x = W[(size_t)(C_ + ka)    * O_ + o];     // Wb[ka][o]
        bD.y = W[(size_t)(C_ + ka + 1)* O_ + o];
        accA = __builtin_amdgcn_wmma_f32_16x16x4_f32(
            /*neg_a=*/false, aF, /*neg_b=*/false, bA,
            /*c_mod=*/(short)0, accA, /*reuse_a=*/false, /*reuse_b=*/false);
        accD = __builtin_amdgcn_wmma_f32_16x16x4_f32(
            false, aF, false, bD, (short)0, accD, false, false);
    }

    const float bo = bias[o];
    #pragma unroll
    for (int i = 0; i < 8; ++i) {
        const int row = rowBase + i + 8 * h;
        const size_t idx = (size_t)(b * N_ + row) * O_ + o;
        const float dv = accD[i];
        dws[idx] = dv;
        u[idx]   = accA[i] - dv + bo;
    }
}

// ---------------------------------------------------------------------------
// Kernel 2: out[b,n,:] = mask[b,n] ? (1/N) * sum_m LN(gelu(u[b,n,:] + d[b,m,:])) : 0
// One block per (b,n). 8 waves, each covers 64 values of m. Lane owns channels
// (2*lane, 2*lane+1) -> float2, coalesced 256B row loads (L2-resident, d[b]=128KB).
// LayerNorm reduction over O=64 done with a single wave32 shuffle tree (sum+sumsq).
// ---------------------------------------------------------------------------
__device__ __forceinline__ float gelu_exact(float v) {
    return 0.5f * v * (1.0f + erff(v * 0.70710678118654752f));
}

__global__ __launch_bounds__(256) void edgeconv_lnmean(
    const float* __restrict__ u,
    const float* __restrict__ dws,
    const int*   __restrict__ mask,   // (B,N)
    const float* __restrict__ gamma,
    const float* __restrict__ beta,
    float* __restrict__ out)          // (B,N,O)
{
    __shared__ float2 part[8 * 32];
    const int tid  = threadIdx.x;
    const int wave = tid >> 5;
    const int lane = tid & 31;
    const int b    = blockIdx.x >> 9;
    const int n    = blockIdx.x & (N_ - 1);
    const size_t row = (size_t)(b * N_ + n);

    if (mask[row] == 0) {             // block-uniform branch; deterministic
        if (tid < 32) {
            float2 z; z.x = 0.0f; z.y = 0.0f;
            *(float2*)(out + row * O_ + 2 * tid) = z;
        }
        return;
    }

    const float2 u2  = *(const float2*)(u     + row * O_ + 2 * lane);
    const float2 g2  = *(const float2*)(gamma + 2 * lane);
    const float2 be2 = *(const float2*)(beta  + 2 * lane);

    const float* dbase = dws + (size_t)b * N_ * O_;
    float2 acc; acc.x = 0.0f; acc.y = 0.0f;

    const int m0 = wave * 64;
    for (int m = m0; m < m0 + 64; ++m) {
        const float2 d2 = *(const float2*)(dbase + (size_t)m * O_ + 2 * lane);
        const float hx = gelu_exact(u2.x + d2.x);
        const float hy = gelu_exact(u2.y + d2.y);
        float s = hx + hy;
        float q = hx * hx + hy * hy;
        #pragma unroll
        for (int off = 16; off > 0; off >>= 1) {
            s += __shfl_xor(s, off, 32);
            q += __shfl_xor(q, off, 32);
        }
        const float mu  = s * (1.0f / 64.0f);
        const float var = q * (1.0f / 64.0f) - mu * mu;
        const float rs  = rsqrtf(var + 1e-5f);
        acc.x += (hx - mu) * rs * g2.x + be2.x;
        acc.y += (hy - mu) * rs * g2.y + be2.y;
    }

    part[wave * 32 + lane] = acc;
    __syncthreads();

    if (tid < 32) {
        float2 res = part[tid];
        #pragma unroll
        for (int w = 1; w < 8; ++w) {
            res.x += part[w * 32 + tid].x;
            res.y += part[w * 32 + tid].y;
        }
        res.x *= (1.0f / (float)N_);
        res.y *= (1.0f / (float)N_);
        *(float2*)(out + row * O_ + 2 * tid) = res;
    }
}

// ---------------------------------------------------------------------------
// Launch: inputs in setup_inputs() order: x, mask, W, b, gamma, beta
// ws layout: [ u : B*N*O f32 ][ d : B*N*O f32 ]  (1 MB total)
// ---------------------------------------------------------------------------
extern "C" void kernel_launch(void* const* d_in, const int* in_sizes, int n_in,
                              void* d_out, int out_size, void* d_ws, size_t ws_size,
                              hipStream_t stream) {
    (void)in_sizes; (void)n_in; (void)out_size; (void)ws_size;
    const float* x     = (const float*)d_in[0];
    const int*   mask  = (const int*)  d_in[1];
    const float* W     = (const float*)d_in[2];
    const float* bias  = (const float*)d_in[3];
    const float* gamma = (const float*)d_in[4];
    const float* beta  = (const float*)d_in[5];
    float* out = (float*)d_out;

    float* u   = (float*)d_ws;
    float* dws = u + (size_t)B_ * N_ * O_;

    // Phase 1: 4 batches * 32 M-tiles = 128 blocks; 4 waves/block = 4 O-tiles.
    edgeconv_gemm<<<B_ * (N_ / 16), 128, 0, stream>>>(x, W, bias, u, dws);
    // Phase 2: one block per (b,n) row.
    edgeconv_lnmean<<<B_ * N_, 256, 0, stream>>>(u, dws, mask, gamma, beta, out);
}